// DigitCaps1D_25280177504310
// MI455X (gfx1250) — compile-verified
//
#include <hip/hip_runtime.h>

// CapsNet dynamic routing on gfx1250.
// u_hat recomputed per routing iteration via V_WMMA_F32_16X16X4_F32:
//   per n:  C(160x64) = W[n](160x8) @ x[:,n,:]^T(8x64)
//   M-tile = one out-capsule o (16 Do rows), N-tile = 16 batch columns,
//   K = 8 = two k-steps of 4.
// C-fragment layout (lane = batch col, vgpr+half = Do row) keeps softmax /
// agreement / s-accumulation lane-local (one shfl_xor(16) for the Do-dot).
// Pass is templated on routing iteration R so the R=0 instantiation carries
// no u/v fragment arrays (stays under 256 VGPRs, no s_set_vgpr_msb churn).

typedef float v2f __attribute__((ext_vector_type(2)));
typedef float v8f __attribute__((ext_vector_type(8)));

#define NTOT 8192
#define BTOT 64
#define OCAP 10
#define DCAP 16
#define DI   8
#define NCHUNKS 64
#define NPER (NTOT / NCHUNKS)   // 128 n's per wave

__device__ __forceinline__ v8f v8f_zero() {
  v8f r;
#pragma unroll
  for (int i = 0; i < 8; ++i) r[i] = 0.0f;
  return r;
}

// One wave (32 lanes) per (batch-tile of 16, chunk of 128 n's).
// R = routing iteration (0,1,2). For R>=1 the coupling logits are
// dot(u_hat, v0 [+ v1]) == dot(u_hat, v0 + v1) -> one combined v fragment.
template <int R>
__global__ __launch_bounds__(32)
void caps_route_pass(const float* __restrict__ x,      // [B, N, Di]
                     const float* __restrict__ W,      // [N, O, Do, Di]
                     float* __restrict__ partials,     // [NCHUNKS, B, O, Do]
                     const float* __restrict__ v0g,    // [B, O, Do]
                     const float* __restrict__ v1g)    // [B, O, Do]
{
  const int lane  = (int)(threadIdx.x & 31u);
  const int half  = lane >> 4;          // which 16-lane half
  const int l16   = lane & 15;
  const int btile = (int)(blockIdx.x & 3u);
  const int chunk = (int)(blockIdx.x >> 2u);
  const int bcol  = btile * 16 + l16;   // this lane's batch column
  const int klo   = half * 2;           // K offset inside a 4-wide k-step

  // s accumulator: same fragment layout as C (10 o-tiles x 8 VGPRs)
  v8f s[OCAP];
#pragma unroll
  for (int o = 0; o < OCAP; ++o) s[o] = v8f_zero();

  // v fragment (R>=1 only): vf[o][k] = v[bcol, o, d = k + 8*half]
  float vf[R >= 1 ? OCAP : 1][8];
  if constexpr (R >= 1) {
#pragma unroll
    for (int o = 0; o < OCAP; ++o)
#pragma unroll
      for (int k = 0; k < 8; ++k) {
        const size_t idx = ((size_t)bcol * OCAP + o) * DCAP + (size_t)(k + 8 * half);
        float t = v0g[idx];
        if constexpr (R == 2) t += v1g[idx];
        vf[o][k] = t;
      }
  } else {
    (void)v0g; (void)v1g;
    vf[0][0] = 0.0f;
  }

  // lane-private contiguous x stream: x[bcol, chunk*NPER + t, :]
  const float* xrow = x + ((size_t)bcol * NTOT + (size_t)chunk * NPER) * DI;

  for (int t = 0; t < NPER; ++t) {
    const int n = chunk * NPER + t;

    // B fragment (x^T, 4x16 per k-step): lane half h holds rows K=2h,2h+1
    v2f bk0, bk1;
    bk0.x = xrow[t * DI + klo];
    bk0.y = xrow[t * DI + klo + 1];
    bk1.x = xrow[t * DI + 4 + klo];
    bk1.y = xrow[t * DI + 4 + klo + 1];

    const float* wn = W + (size_t)n * (OCAP * DCAP * DI);

    if constexpr (R == 0) {
      // uniform coupling (softmax of zeros): s[o] += 0.1 * u[o], no u array
#pragma unroll
      for (int o = 0; o < OCAP; ++o) {
        const float* wrow = wn + (size_t)(o * DCAP + l16) * DI;
        v2f ak0, ak1;
        ak0.x = wrow[klo];     ak0.y = wrow[klo + 1];
        ak1.x = wrow[4 + klo]; ak1.y = wrow[4 + klo + 1];
        v8f acc = v8f_zero();
        acc = __builtin_amdgcn_wmma_f32_16x16x4_f32(false, ak0, false, bk0,
                                                    (short)0, acc, false, false);
        acc = __builtin_amdgcn_wmma_f32_16x16x4_f32(false, ak1, false, bk1,
                                                    (short)0, acc, false, false);
#pragma unroll
        for (int k = 0; k < 8; ++k) s[o][k] += 0.1f * acc[k];
      }
    } else {
      v8f u[OCAP];
#pragma unroll
      for (int o = 0; o < OCAP; ++o) {
        // A fragment (16x4): lane l16 holds row M=d=l16; half h -> K=2h,2h+1
        const float* wrow = wn + (size_t)(o * DCAP + l16) * DI;
        v2f ak0, ak1;
        ak0.x = wrow[klo];     ak0.y = wrow[klo + 1];
        ak1.x = wrow[4 + klo]; ak1.y = wrow[4 + klo + 1];
        v8f acc = v8f_zero();
        acc = __builtin_amdgcn_wmma_f32_16x16x4_f32(false, ak0, false, bk0,
                                                    (short)0, acc, false, false);
        acc = __builtin_amdgcn_wmma_f32_16x16x4_f32(false, ak1, false, bk1,
                                                    (short)0, acc, false, false);
        u[o] = acc;
      }

      // routing coefficients c[b,n,o] (lane-local, b fixed per lane)
      float c[OCAP];
      float logit[OCAP];
      float m = -3.4e38f;
#pragma unroll
      for (int o = 0; o < OCAP; ++o) {
        float p = 0.0f;
#pragma unroll
        for (int k = 0; k < 8; ++k) p += u[o][k] * vf[o][k];
        p += __shfl_xor(p, 16, 32);   // partner half holds the other 8 Do's
        logit[o] = p;
        m = fmaxf(m, p);
      }
      float sum = 0.0f;
#pragma unroll
      for (int o = 0; o < OCAP; ++o) { c[o] = __expf(logit[o] - m); sum += c[o]; }
      const float inv = 1.0f / sum;
#pragma unroll
      for (int o = 0; o < OCAP; ++o)
#pragma unroll
        for (int k = 0; k < 8; ++k) s[o][k] += (c[o] * inv) * u[o][k];
    }
  }

  // spill this wave's partial: partials[chunk][b][o][d]
  float* pp = partials + ((size_t)chunk * BTOT + bcol) * (OCAP * DCAP);
#pragma unroll
  for (int o = 0; o < OCAP; ++o)
#pragma unroll
    for (int k = 0; k < 8; ++k)
      pp[o * DCAP + (k + 8 * half)] = s[o][k];
}

// Deterministic reduction over NCHUNKS + squash.
// NOTE: reference squash sums the squared norm over the OUT-CAPSULE axis
// (faithful copy of the original bug): norm[b,d] = sum_o s[b,o,d]^2.
__global__ __launch_bounds__(OCAP * DCAP)
void caps_reduce_squash(const float* __restrict__ partials,  // [NCHUNKS,B,O,Do]
                        float* __restrict__ vout)            // [B, O, Do]
{
  const int b = (int)blockIdx.x;
  const int t = (int)threadIdx.x;      // t = o*16 + d
  const int d = t & 15;

  float sv = 0.0f;
  for (int c = 0; c < NCHUNKS; ++c)
    sv += partials[((size_t)c * BTOT + b) * (OCAP * DCAP) + t];

  __shared__ float sh[OCAP * DCAP];
  sh[t] = sv;
  __syncthreads();

  float norm = 0.0f;
#pragma unroll
  for (int oo = 0; oo < OCAP; ++oo) { const float q = sh[oo * DCAP + d]; norm += q * q; }
  const float scale = norm / (1.0f + norm) * rsqrtf(norm + 1e-9f);
  vout[(size_t)b * (OCAP * DCAP) + t] = scale * sv;
}

extern "C" void kernel_launch(void* const* d_in, const int* in_sizes, int n_in,
                              void* d_out, int out_size, void* d_ws, size_t ws_size,
                              hipStream_t stream) {
  (void)in_sizes; (void)n_in; (void)out_size; (void)ws_size;

  const float* x = (const float*)d_in[0];   // [64, 8192, 8]
  const float* W = (const float*)d_in[1];   // [1, 8192, 10, 16, 8]
  float* out = (float*)d_out;               // [64, 10, 16]

  // workspace layout (floats): partials | v0 | v1   (~2.7 MB total)
  float* ws       = (float*)d_ws;
  float* partials = ws;
  float* v0       = ws + (size_t)NCHUNKS * BTOT * OCAP * DCAP;
  float* v1       = v0 + (size_t)BTOT * OCAP * DCAP;

  const dim3 gp(NCHUNKS * 4);      // (chunk, btile)
  const dim3 bp(32);               // one wave
  const dim3 gr(BTOT);
  const dim3 br(OCAP * DCAP);      // 160 = 5 waves

  // r=0: uniform coupling -> s0 -> v0
  caps_route_pass<0><<<gp, bp, 0, stream>>>(x, W, partials, v0, v1);
  caps_reduce_squash<<<gr, br, 0, stream>>>(partials, v0);
  // r=1: logits = u.v0 -> s1 -> v1
  caps_route_pass<1><<<gp, bp, 0, stream>>>(x, W, partials, v0, v1);
  caps_reduce_squash<<<gr, br, 0, stream>>>(partials, v1);
  // r=2: logits = u.(v0+v1) -> s2 -> squash -> output
  caps_route_pass<2><<<gp, bp, 0, stream>>>(x, W, partials, v0, v1);
  caps_reduce_squash<<<gr, br, 0, stream>>>(partials, out);
}